// SequenceStateUpdater_76897094468326
// MI455X (gfx1250) — compile-verified
//
#include <hip/hip_runtime.h>

#define BATCH     8192
#define N_NODES   100000
#define DIM       256          // STATE_DIM == MSG_DIM
#define ROWS      32           // batch rows per workgroup (two M-tiles)
#define WAVES     16           // one wave per 16-wide column group (16*16 = 256 cols)

typedef __attribute__((ext_vector_type(2))) float v2f;
typedef __attribute__((ext_vector_type(8))) float v8f;

#define WMMA_F32(a, b, c) \
    __builtin_amdgcn_wmma_f32_16x16x4_f32(false, (a), false, (b), (short)0, (c), false, false)

__device__ __forceinline__ float sigmoidf_(float x) {
    return 1.0f / (1.0f + expf(-x));
}

// Scatter the last-update tables.  src then dst on the shared user table is
// order-independent under max (values >= 0), so an integer-bit atomicMax on the
// float (monotone for non-negative floats) reproduces the sequential semantics.
__global__ void lu_update_kernel(const int* __restrict__ src_ids, const float* __restrict__ src_ts,
                                 const int* __restrict__ dst_ids, const float* __restrict__ dst_ts,
                                 const int* __restrict__ cas_ids, const float* __restrict__ cas_ts,
                                 float* __restrict__ user_lu, float* __restrict__ cas_lu) {
    int i = blockIdx.x * blockDim.x + threadIdx.x;
    if (i < BATCH) {
        atomicMax((int*)(user_lu + src_ids[i]), __float_as_int(src_ts[i]));
        atomicMax((int*)(user_lu + dst_ids[i]), __float_as_int(dst_ts[i]));
        cas_lu[cas_ids[i]] = cas_ts[i];
    }
}

// One GRU cell: gather H, gi = X@W_ih^T, gh = H@W_hh^T via f32 WMMA with a
// software-pipelined K loop (prefetch next-K operands while current WMMAs run),
// fused pointwise GRU, scatter h' into state_out (pre-copied from state_in).
__global__ __launch_bounds__(WAVES * 32)
void gru_scatter_kernel(const int*   __restrict__ ids,
                        const float* __restrict__ msgs,      // [BATCH, 256]
                        const float* __restrict__ state_in,  // [N_NODES, 256]
                        const float* __restrict__ W_ih,      // [768, 256] row-major
                        const float* __restrict__ W_hh,      // [768, 256] row-major
                        const float* __restrict__ b_ih,      // [768]
                        const float* __restrict__ b_hh,      // [768]
                        float*       __restrict__ state_out) // [N_NODES, 256]
{
    __shared__ float Xs[ROWS * DIM];   // 32 KB
    __shared__ float Hs[ROWS * DIM];   // 32 KB

    const int tid     = threadIdx.x;
    const int rowBase = blockIdx.x * ROWS;

    // Coalesced stage of X tile and gathered H tile: 2048 float4 each.
    #pragma unroll
    for (int j = 0; j < 4; ++j) {
        int i  = tid + j * (WAVES * 32);   // 0..2047
        int r  = i >> 6;                   // row within tile (64 float4 per row)
        int c4 = i & 63;                   // float4 column
        ((float4*)Xs)[i] = ((const float4*)(msgs + (size_t)(rowBase + r) * DIM))[c4];
        int node = ids[rowBase + r];
        ((float4*)Hs)[i] = ((const float4*)(state_in + (size_t)node * DIM))[c4];
    }
    __syncthreads();

    const int lane = tid & 31;
    const int wave = tid >> 5;       // column group g: columns 16g .. 16g+15
    const int half = lane >> 4;      // selects K pair {0,1} vs {2,3} per ISA layout
    const int l15  = lane & 15;
    const int koff = half * 2;

    // Per-lane A-operand bases (LDS), one per M-tile.
    const float* xs0 = Xs + l15 * DIM + koff;
    const float* xs1 = Xs + (16 + l15) * DIM + koff;
    const float* hs0 = Hs + l15 * DIM + koff;
    const float* hs1 = Hs + (16 + l15) * DIM + koff;

    // Per-lane B-operand bases (global weights, K-contiguous pairs).
    const float* wr_i = W_ih + (size_t)(        wave * 16 + l15) * DIM + koff;
    const float* wz_i = W_ih + (size_t)(DIM   + wave * 16 + l15) * DIM + koff;
    const float* wn_i = W_ih + (size_t)(2*DIM + wave * 16 + l15) * DIM + koff;
    const float* wr_h = W_hh + (size_t)(        wave * 16 + l15) * DIM + koff;
    const float* wz_h = W_hh + (size_t)(DIM   + wave * 16 + l15) * DIM + koff;
    const float* wn_h = W_hh + (size_t)(2*DIM + wave * 16 + l15) * DIM + koff;

    v8f a_ir0 = {}; v8f a_iz0 = {}; v8f a_in0 = {};
    v8f a_ir1 = {}; v8f a_iz1 = {}; v8f a_in1 = {};
    v8f a_hr0 = {}; v8f a_hz0 = {}; v8f a_hn0 = {};
    v8f a_hr1 = {}; v8f a_hz1 = {}; v8f a_hn1 = {};

    // Pipeline prologue: operands for k = 0.
    v2f ax0 = *(const v2f*)(xs0);  v2f ax1 = *(const v2f*)(xs1);
    v2f ah0 = *(const v2f*)(hs0);  v2f ah1 = *(const v2f*)(hs1);
    v2f bri = *(const v2f*)(wr_i); v2f bzi = *(const v2f*)(wz_i); v2f bni = *(const v2f*)(wn_i);
    v2f brh = *(const v2f*)(wr_h); v2f bzh = *(const v2f*)(wz_h); v2f bnh = *(const v2f*)(wn_h);

    for (int k = 0; k < DIM; k += 4) {
        // Prefetch next K-step operands (wraps to 0 on the last iteration —
        // harmless redundant load, keeps the loop branch-free and uniform).
        const int kn = (k + 4) & (DIM - 1);
        v2f n_ax0 = *(const v2f*)(xs0 + kn);  v2f n_ax1 = *(const v2f*)(xs1 + kn);
        v2f n_ah0 = *(const v2f*)(hs0 + kn);  v2f n_ah1 = *(const v2f*)(hs1 + kn);
        v2f n_bri = *(const v2f*)(wr_i + kn); v2f n_bzi = *(const v2f*)(wz_i + kn);
        v2f n_bni = *(const v2f*)(wn_i + kn); v2f n_brh = *(const v2f*)(wr_h + kn);
        v2f n_bzh = *(const v2f*)(wz_h + kn); v2f n_bnh = *(const v2f*)(wn_h + kn);

        // 12 WMMAs on the already-resident operands; prefetch stays in flight.
        a_ir0 = WMMA_F32(ax0, bri, a_ir0);
        a_ir1 = WMMA_F32(ax1, bri, a_ir1);
        a_iz0 = WMMA_F32(ax0, bzi, a_iz0);
        a_iz1 = WMMA_F32(ax1, bzi, a_iz1);
        a_in0 = WMMA_F32(ax0, bni, a_in0);
        a_in1 = WMMA_F32(ax1, bni, a_in1);
        a_hr0 = WMMA_F32(ah0, brh, a_hr0);
        a_hr1 = WMMA_F32(ah1, brh, a_hr1);
        a_hz0 = WMMA_F32(ah0, bzh, a_hz0);
        a_hz1 = WMMA_F32(ah1, bzh, a_hz1);
        a_hn0 = WMMA_F32(ah0, bnh, a_hn0);
        a_hn1 = WMMA_F32(ah1, bnh, a_hn1);

        ax0 = n_ax0; ax1 = n_ax1; ah0 = n_ah0; ah1 = n_ah1;
        bri = n_bri; bzi = n_bzi; bni = n_bni;
        brh = n_brh; bzh = n_bzh; bnh = n_bnh;
    }

    // Pointwise GRU + scatter.  C/D layout: VGPR v -> row m = v + 8*half,
    // lanes 0-15 / 16-31 -> column = l15 within the tile.
    const int   col  = wave * 16 + l15;
    const float bi_r = b_ih[col],           bh_r = b_hh[col];
    const float bi_z = b_ih[DIM + col],     bh_z = b_hh[DIM + col];
    const float bi_n = b_ih[2*DIM + col],   bh_n = b_hh[2*DIM + col];

    #pragma unroll
    for (int v = 0; v < 8; ++v) {          // M-tile 0: rows 0..15
        int   m = v + half * 8;
        float r = sigmoidf_(a_ir0[v] + bi_r + a_hr0[v] + bh_r);
        float z = sigmoidf_(a_iz0[v] + bi_z + a_hz0[v] + bh_z);
        float n = tanhf(a_in0[v] + bi_n + r * (a_hn0[v] + bh_n));
        float h = Hs[m * DIM + col];
        int   node = ids[rowBase + m];
        state_out[(size_t)node * DIM + col] = (1.0f - z) * n + z * h;
    }
    #pragma unroll
    for (int v = 0; v < 8; ++v) {          // M-tile 1: rows 16..31
        int   m = 16 + v + half * 8;
        float r = sigmoidf_(a_ir1[v] + bi_r + a_hr1[v] + bh_r);
        float z = sigmoidf_(a_iz1[v] + bi_z + a_hz1[v] + bh_z);
        float n = tanhf(a_in1[v] + bi_n + r * (a_hn1[v] + bh_n));
        float h = Hs[m * DIM + col];
        int   node = ids[rowBase + m];
        state_out[(size_t)node * DIM + col] = (1.0f - z) * n + z * h;
    }
}

extern "C" void kernel_launch(void* const* d_in, const int* in_sizes, int n_in,
                              void* d_out, int out_size, void* d_ws, size_t ws_size,
                              hipStream_t stream) {
    (void)in_sizes; (void)n_in; (void)out_size; (void)d_ws; (void)ws_size;

    const int*   src_ids   = (const int*)  d_in[0];
    const int*   dst_ids   = (const int*)  d_in[1];
    const int*   cas_ids   = (const int*)  d_in[2];
    const float* src_msgs  = (const float*)d_in[3];
    const float* dst_msgs  = (const float*)d_in[4];
    const float* cas_msgs  = (const float*)d_in[5];
    const float* src_ts    = (const float*)d_in[6];
    const float* dst_ts    = (const float*)d_in[7];
    const float* cas_ts    = (const float*)d_in[8];
    const float* user_lu   = (const float*)d_in[9];
    const float* cas_lu    = (const float*)d_in[10];
    const float* src_state = (const float*)d_in[11];
    const float* dst_state = (const float*)d_in[12];
    const float* cas_state = (const float*)d_in[13];
    // params_src / params_dst / params_cas, dict order: W_ih, W_hh, b_ih, b_hh
    const float* Wih_s = (const float*)d_in[14]; const float* Whh_s = (const float*)d_in[15];
    const float* bih_s = (const float*)d_in[16]; const float* bhh_s = (const float*)d_in[17];
    const float* Wih_d = (const float*)d_in[18]; const float* Whh_d = (const float*)d_in[19];
    const float* bih_d = (const float*)d_in[20]; const float* bhh_d = (const float*)d_in[21];
    const float* Wih_c = (const float*)d_in[22]; const float* Whh_c = (const float*)d_in[23];
    const float* bih_c = (const float*)d_in[24]; const float* bhh_c = (const float*)d_in[25];

    float* out_src = (float*)d_out;
    float* out_dst = out_src + (size_t)N_NODES * DIM;
    float* out_cas = out_dst + (size_t)N_NODES * DIM;
    float* out_ulu = out_cas + (size_t)N_NODES * DIM;
    float* out_clu = out_ulu + N_NODES;

    const size_t stateBytes = (size_t)N_NODES * DIM * sizeof(float);
    const size_t luBytes    = (size_t)N_NODES * sizeof(float);

    // Full-table copies (the functional output includes all untouched rows).
    hipMemcpyAsync(out_src, src_state, stateBytes, hipMemcpyDeviceToDevice, stream);
    hipMemcpyAsync(out_dst, dst_state, stateBytes, hipMemcpyDeviceToDevice, stream);
    hipMemcpyAsync(out_cas, cas_state, stateBytes, hipMemcpyDeviceToDevice, stream);
    hipMemcpyAsync(out_ulu, user_lu,   luBytes,    hipMemcpyDeviceToDevice, stream);
    hipMemcpyAsync(out_clu, cas_lu,    luBytes,    hipMemcpyDeviceToDevice, stream);

    lu_update_kernel<<<(BATCH + 255) / 256, 256, 0, stream>>>(
        src_ids, src_ts, dst_ids, dst_ts, cas_ids, cas_ts, out_ulu, out_clu);

    dim3 grid(BATCH / ROWS);
    dim3 block(WAVES * 32);
    gru_scatter_kernel<<<grid, block, 0, stream>>>(
        src_ids, src_msgs, src_state, Wih_s, Whh_s, bih_s, bhh_s, out_src);
    gru_scatter_kernel<<<grid, block, 0, stream>>>(
        dst_ids, dst_msgs, dst_state, Wih_d, Whh_d, bih_d, bhh_d, out_dst);
    gru_scatter_kernel<<<grid, block, 0, stream>>>(
        cas_ids, cas_msgs, cas_state, Wih_c, Whh_c, bih_c, bhh_c, out_cas);
}